// HeteroMessagePassingLayer_24696061952069
// MI455X (gfx1250) — compile-verified
//
#include <hip/hip_runtime.h>

#define D   128
#define EPS 1e-5f

typedef __attribute__((ext_vector_type(2))) float v2f;
typedef __attribute__((ext_vector_type(8))) float v8f;

// ---------------------------------------------------------------------------
// Zero a float region (grid-stride).
// ---------------------------------------------------------------------------
__global__ __launch_bounds__(256) void zero_f32(float* __restrict__ p, long n) {
  long i = (long)blockIdx.x * blockDim.x + threadIdx.x;
  long stride = (long)gridDim.x * blockDim.x;
  for (; i < n; i += stride) p[i] = 0.0f;
}

// ---------------------------------------------------------------------------
// Pack weight panels:
//   BpA[n][k] (k<384): k<128 -> W_l_aa[n][k]; k<256 -> W_l_ba[n][k-128];
//                      else  -> W_r_aa[n][k-256] + W_r_ba[n][k-256]
//   BpB[n][k] (k<256): k<128 -> W_l_ab[n][k];  else -> W_r_ab[n][k-128]
//   biasA = b_aa + b_ba ; biasB = b_ab
// (C[m][n] = sum_k A[m][k] * W[n][k]  since the reference does  x @ W.T)
// ---------------------------------------------------------------------------
__global__ __launch_bounds__(256) void pack_weights(
    const float* __restrict__ Wl_aa, const float* __restrict__ Wr_aa,
    const float* __restrict__ b_aa,  const float* __restrict__ Wl_ba,
    const float* __restrict__ Wr_ba, const float* __restrict__ b_ba,
    const float* __restrict__ Wl_ab, const float* __restrict__ Wr_ab,
    const float* __restrict__ b_ab,
    float* __restrict__ BpA, float* __restrict__ BpB,
    float* __restrict__ biasA, float* __restrict__ biasB) {
  int i = blockIdx.x * blockDim.x + threadIdx.x;
  if (i < D * 384) {
    int n = i / 384, k = i % 384;
    float v;
    if (k < 128)       v = Wl_aa[n * D + k];
    else if (k < 256)  v = Wl_ba[n * D + (k - 128)];
    else               v = Wr_aa[n * D + (k - 256)] + Wr_ba[n * D + (k - 256)];
    BpA[(size_t)n * 384 + k] = v;
  }
  if (i < D * 256) {
    int n = i / 256, k = i % 256;
    BpB[(size_t)n * 256 + k] = (k < 128) ? Wl_ab[n * D + k]
                                         : Wr_ab[n * D + (k - 128)];
  }
  if (i < D) {
    biasA[i] = b_aa[i] + b_ba[i];
    biasB[i] = b_ab[i];
  }
}

// ---------------------------------------------------------------------------
// Edge scatter: one wave32 per edge. Lane l accumulates 4 consecutive floats
// of the source row into sum[dst] (global_atomic_add_f32); lane 0 counts.
// ---------------------------------------------------------------------------
__global__ __launch_bounds__(256) void scatter_mean_accum(
    const float* __restrict__ xsrc, const int* __restrict__ srci,
    const int* __restrict__ dsti, float* __restrict__ sum,
    float* __restrict__ cnt, int nedges) {
  int gid  = blockIdx.x * blockDim.x + threadIdx.x;
  int lane = gid & 31;
  int e    = gid >> 5;
  if (e >= nedges) return;
  int s = srci[e];
  int d = dsti[e];
  const float4 v = *(const float4*)(xsrc + (size_t)s * D + lane * 4);
  float* o = sum + (size_t)d * D + lane * 4;
  atomicAdd(o + 0, v.x);
  atomicAdd(o + 1, v.y);
  atomicAdd(o + 2, v.z);
  atomicAdd(o + 3, v.w);
  if (lane == 0) atomicAdd(cnt + d, 1.0f);
}

// ---------------------------------------------------------------------------
// Fused (concat-K GEMM) + bias + LayerNorm + ReLU.
//   A = [ s0*diag(1/cnt0) | s1*diag(1/cnt1) | s2 ]   (NSRC sources, K=NSRC*128)
//   B = packed weight panel [n][k]
// One wave computes a 16-row strip across the full N=128 output
// (8 x 16x16 f32 WMMA accumulators), then does the LayerNorm in-wave.
//
// WMMA f32 16x16x4 operand layout (ISA 7.12.2):
//   A v2f: lane<16 -> {A[M=l][k0], A[M=l][k0+1]} ; lane>=16 -> k0+2, k0+3
//   B v2f: lane<16 -> {B[k0][N=l], B[k0+1][N=l]} ; lane>=16 -> k0+2, k0+3
//   C v8f: element r -> C[M = r + 8*half][N = lane&15]
// ---------------------------------------------------------------------------
template <int NSRC, int KTOT>
__global__ __launch_bounds__(256) void sage_gemm_ln(
    const float* __restrict__ s0, const float* __restrict__ c0,
    const float* __restrict__ s1, const float* __restrict__ c1,
    const float* __restrict__ s2, const float* __restrict__ c2,
    const float* __restrict__ Bp, const float* __restrict__ bias,
    const float* __restrict__ gamma, const float* __restrict__ beta,
    float* __restrict__ out, int nrows) {
  constexpr int LSTR = KTOT + 2;       // pad 2 dwords -> conflict-free ds_load_b64
  extern __shared__ float sB[];        // [128][LSTR]
  const int tid  = threadIdx.x;
  const int wid  = tid >> 5;
  const int lane = tid & 31;

  // Cooperative stage of the weight panel into LDS.
  for (int n = wid; n < D; n += 8)
    for (int k = lane * 2; k < KTOT; k += 64)
      *(v2f*)(sB + n * LSTR + k) = *(const v2f*)(Bp + (size_t)n * KTOT + k);
  __syncthreads();

  const int strip = blockIdx.x * 8 + wid;
  if (strip * 16 >= nrows) return;     // wave-uniform; EXEC stays all-ones
  const int rowbase = strip * 16;
  const int half = lane >> 4;
  const int l16  = lane & 15;
  const int rA   = rowbase + l16;      // A-fragment row for this lane

  v8f zero = {0.f, 0.f, 0.f, 0.f, 0.f, 0.f, 0.f, 0.f};
  v8f acc[8] = {zero, zero, zero, zero, zero, zero, zero, zero};

  const float* srcs[3] = {s0, s1, s2};
  const float* cnts[3] = {c0, c1, c2};

  for (int s = 0; s < NSRC; s++) {
    const float* S = srcs[s] + (size_t)rA * D;
    float scale = 1.0f;
    if (cnts[s]) scale = 1.0f / fmaxf(cnts[s][rA], 1.0f);  // fold mean into A
    const int kbase = s * D;
#pragma unroll 4
    for (int c = 0; c < 32; c++) {               // K chunks of 4 within source
      v2f a = *(const v2f*)(S + 4 * c + 2 * half);
      a *= scale;
      const float* bptr = sB + (kbase + 4 * c + 2 * half);
#pragma unroll
      for (int t = 0; t < 8; t++) {              // 8 column tiles -> N=128
        v2f b = *(const v2f*)(bptr + (t * 16 + l16) * LSTR);
        acc[t] = __builtin_amdgcn_wmma_f32_16x16x4_f32(
            false, a, false, b, (short)0, acc[t], false, false);
      }
    }
  }

  // bias (column-broadcast per tile)
#pragma unroll
  for (int t = 0; t < 8; t++) {
    float bb = bias[t * 16 + l16];
#pragma unroll
    for (int r = 0; r < 8; r++) acc[t][r] += bb;
  }

  // LayerNorm stats: row (rowbase + r + 8*half) lives in v8f element r,
  // spread across this 16-lane half (8 columns per lane).
  float mu[8], rstd[8];
#pragma unroll
  for (int r = 0; r < 8; r++) {
    float sv = 0.f, qv = 0.f;
#pragma unroll
    for (int t = 0; t < 8; t++) {
      float x = acc[t][r];
      sv += x;
      qv += x * x;
    }
    for (int m = 1; m < 16; m <<= 1) {
      sv += __shfl_xor(sv, m, 16);
      qv += __shfl_xor(qv, m, 16);
    }
    float m1  = sv * (1.0f / D);
    float var = qv * (1.0f / D) - m1 * m1;
    mu[r]   = m1;
    rstd[r] = rsqrtf(var + EPS);
  }

  // gamma/beta + ReLU + store
#pragma unroll
  for (int t = 0; t < 8; t++) {
    int n = t * 16 + l16;
    float g  = gamma[n];
    float bt = beta[n];
#pragma unroll
    for (int r = 0; r < 8; r++) {
      int row = rowbase + half * 8 + r;
      float y = (acc[t][r] - mu[r]) * rstd[r] * g + bt;
      out[(size_t)row * D + n] = fmaxf(y, 0.0f);
    }
  }
}

// ---------------------------------------------------------------------------
extern "C" void kernel_launch(void* const* d_in, const int* in_sizes, int n_in,
                              void* d_out, int out_size, void* d_ws,
                              size_t ws_size, hipStream_t stream) {
  const float* x_A    = (const float*)d_in[0];
  const float* x_B    = (const float*)d_in[1];
  const int*   ei_aa  = (const int*)d_in[2];
  const int*   ei_ba  = (const int*)d_in[3];
  const int*   ei_ab  = (const int*)d_in[4];
  const float* W_l_aa = (const float*)d_in[5];
  const float* W_r_aa = (const float*)d_in[6];
  const float* b_aa   = (const float*)d_in[7];
  const float* W_l_ba = (const float*)d_in[8];
  const float* W_r_ba = (const float*)d_in[9];
  const float* b_ba   = (const float*)d_in[10];
  const float* W_l_ab = (const float*)d_in[11];
  const float* W_r_ab = (const float*)d_in[12];
  const float* b_ab   = (const float*)d_in[13];
  const float* gamma  = (const float*)d_in[14];
  const float* beta   = (const float*)d_in[15];
  float* out = (float*)d_out;

  const int nA   = in_sizes[0] / D;
  const int nB   = in_sizes[1] / D;
  const int E_aa = in_sizes[2] / 2;
  const int E_ba = in_sizes[3] / 2;
  const int E_ab = in_sizes[4] / 2;

  // Workspace layout (floats)
  float* ws     = (float*)d_ws;
  float* sum_aa = ws;
  float* sum_ba = sum_aa + (size_t)nA * D;
  float* sum_ab = sum_ba + (size_t)nA * D;
  float* cnt_aa = sum_ab + (size_t)nB * D;
  float* cnt_ba = cnt_aa + nA;
  float* cnt_ab = cnt_ba + nA;
  float* BpA    = cnt_ab + nB;
  float* BpB    = BpA + (size_t)D * 384;
  float* biasA  = BpB + (size_t)D * 256;
  float* biasB  = biasA + D;

  // 1) zero accumulators (sums + counts are contiguous)
  long nzero = (long)2 * nA * D + (long)nB * D + 2L * nA + nB;
  zero_f32<<<4096, 256, 0, stream>>>(ws, nzero);

  // 2) pack weight panels / bias sums
  pack_weights<<<(D * 384 + 255) / 256, 256, 0, stream>>>(
      W_l_aa, W_r_aa, b_aa, W_l_ba, W_r_ba, b_ba, W_l_ab, W_r_ab, b_ab,
      BpA, BpB, biasA, biasB);

  // 3) edge scatter (one wave per edge)
  {
    int blk;
    blk = (int)(((long)E_aa * 32 + 255) / 256);
    scatter_mean_accum<<<blk, 256, 0, stream>>>(x_A, ei_aa, ei_aa + E_aa,
                                                sum_aa, cnt_aa, E_aa);
    blk = (int)(((long)E_ba * 32 + 255) / 256);
    scatter_mean_accum<<<blk, 256, 0, stream>>>(x_B, ei_ba, ei_ba + E_ba,
                                                sum_ba, cnt_ba, E_ba);
    blk = (int)(((long)E_ab * 32 + 255) / 256);
    scatter_mean_accum<<<blk, 256, 0, stream>>>(x_A, ei_ab, ei_ab + E_ab,
                                                sum_ab, cnt_ab, E_ab);
  }

  // 4) fused GEMM + LayerNorm + ReLU
  const int ldsA = (384 + 2) * D * 4;  // 197,632 B
  const int ldsB = (256 + 2) * D * 4;  // 132,096 B
  hipFuncSetAttribute(reinterpret_cast<const void*>(&sage_gemm_ln<3, 384>),
                      hipFuncAttributeMaxDynamicSharedMemorySize, ldsA);
  hipFuncSetAttribute(reinterpret_cast<const void*>(&sage_gemm_ln<2, 256>),
                      hipFuncAttributeMaxDynamicSharedMemorySize, ldsB);

  int stripsA = (nA + 15) / 16;
  int stripsB = (nB + 15) / 16;
  sage_gemm_ln<3, 384><<<(stripsA + 7) / 8, 256, ldsA, stream>>>(
      sum_aa, cnt_aa, sum_ba, cnt_ba, x_A, nullptr, BpA, biasA, gamma, beta,
      out, nA);
  sage_gemm_ln<2, 256><<<(stripsB + 7) / 8, 256, ldsB, stream>>>(
      sum_ab, cnt_ab, x_B, nullptr, nullptr, nullptr, BpB, biasB, gamma, beta,
      out + (size_t)nA * D, nB);
}